// indi_GCNwithJK_1623497638167
// MI455X (gfx1250) — compile-verified
//
#include <hip/hip_runtime.h>
#include <hip/hip_bf16.h>

typedef __attribute__((ext_vector_type(2))) float v2f;
typedef __attribute__((ext_vector_type(8))) float v8f;

#define K_DIM 128
#define BN_EPS 1e-5f

// ---------------------------------------------------------------------------
// Degree / normalization kernels
// ---------------------------------------------------------------------------
__global__ void deg_init_kernel(float* __restrict__ deg, int n) {
    int i = blockIdx.x * blockDim.x + threadIdx.x;
    if (i < n) deg[i] = 1.0f;   // self-loop contributes 1
}

__global__ void deg_count_kernel(const int* __restrict__ dst, float* __restrict__ deg, int e) {
    int i = blockIdx.x * blockDim.x + threadIdx.x;
    if (i < e) atomicAdd(&deg[dst[i]], 1.0f);
}

__global__ void deg_rsqrt_kernel(float* __restrict__ deg, int n) {
    int i = blockIdx.x * blockDim.x + threadIdx.x;
    if (i < n) deg[i] = rsqrtf(deg[i]);
}

// agg[i,:] = xw[i,:] * disq[i]^2   (self-loop term, also zero-initializes agg)
__global__ void selfloop_init_kernel(const float* __restrict__ xw,
                                     const float* __restrict__ disq,
                                     float* __restrict__ agg, int n) {
    int idx = blockIdx.x * blockDim.x + threadIdx.x;   // over n*32 float4 groups
    if (idx >= n * 32) return;
    int row = idx >> 5;
    float s = disq[row];
    s = s * s;
    float4 v = ((const float4*)xw)[idx];
    v.x *= s; v.y *= s; v.z *= s; v.w *= s;
    ((float4*)agg)[idx] = v;
}

// one wave per edge: agg[dst,:] += xw[src,:] * disq[src]*disq[dst]
__global__ void scatter_kernel(const float* __restrict__ xw,
                               const int* __restrict__ src,
                               const int* __restrict__ dst,
                               const float* __restrict__ disq,
                               float* __restrict__ agg, int e) {
    int w = blockIdx.x * (blockDim.x >> 5) + (threadIdx.x >> 5);
    if (w >= e) return;
    int lane = threadIdx.x & 31;
    int s = src[w];
    int d = dst[w];
    float nrm = disq[s] * disq[d];
    const float4 v = *(const float4*)(xw + (size_t)s * K_DIM + lane * 4);
    float* ap = agg + (size_t)d * K_DIM + lane * 4;
    atomicAdd(ap + 0, v.x * nrm);
    atomicAdd(ap + 1, v.y * nrm);
    atomicAdd(ap + 2, v.z * nrm);
    atomicAdd(ap + 3, v.w * nrm);
}

// ---------------------------------------------------------------------------
// Fused GEMM: Y[m,n] = act( bn( sum_k X[m,k]*W[n,k] + bias[n] ) )
//   JK_MODE 0: none   1: JK = Y (copy)   2: JK = max(JK, Y)
// 256 threads = 8 waves; each wave computes a 16-row x NOUT tile with
// V_WMMA_F32_16X16X4_F32 (fp32 matrix path, K stepped 4 at a time).
// W staged in LDS with padded stride 132 (bank-conflict-free column reads).
// ---------------------------------------------------------------------------
template <int NOUT, bool RELU, int JK_MODE>
__global__ __launch_bounds__(256) void gemm_epilogue_kernel(
    const float* __restrict__ X,
    const float* __restrict__ W,         // [NOUT, 128] row-major
    const float* __restrict__ bias,      // nullptr -> no bias
    const float* __restrict__ bn_gamma,  // nullptr -> no BN
    const float* __restrict__ bn_beta,
    float* __restrict__ Y,               // [nrows, NOUT]
    float* __restrict__ JK,              // [nrows, 128] (modes 1/2)
    int nrows) {
    constexpr int NT  = (NOUT + 15) / 16;   // column tiles
    constexpr int LDW = 132;                // padded LDS stride (floats)
    extern __shared__ float lds_w[];        // NT*16 * LDW floats

    const int tid = threadIdx.x;

    // Cooperative load of W into LDS (rows beyond NOUT zero-padded)
    for (int idx = tid; idx < NT * 16 * 32; idx += 256) {
        int row = idx >> 5;
        int c4  = (idx & 31) << 2;
        float4 v = make_float4(0.f, 0.f, 0.f, 0.f);
        if (row < NOUT) v = *(const float4*)(W + row * K_DIM + c4);
        *(float4*)(&lds_w[row * LDW + c4]) = v;
    }
    __syncthreads();

    const int wave = tid >> 5;
    const int lane = tid & 31;
    const int half = lane >> 4;  // 0 or 1
    const int lr   = lane & 15;
    const int m0   = blockIdx.x * 128 + wave * 16;
    if (m0 >= nrows) return;     // wave-uniform; EXEC stays all-ones for WMMA

    // A fragment source: lane holds row m0+lr, K components (k0+2*half, +1)
    const float* xrow = X + (size_t)(m0 + lr) * K_DIM + half * 2;

    v8f acc[NT] = {};

#pragma unroll 4
    for (int kk = 0; kk < 32; ++kk) {
        v2f a = *(const v2f*)(xrow + kk * 4);
        const float* wb = &lds_w[lr * LDW + kk * 4 + half * 2];
#pragma unroll
        for (int j = 0; j < NT; ++j) {
            v2f b = *(const v2f*)(wb + j * 16 * LDW);
            acc[j] = __builtin_amdgcn_wmma_f32_16x16x4_f32(
                false, a, false, b, (short)0, acc[j], false, false);
        }
    }

    // Epilogue: bias + eval-mode BN + ReLU + JumpingKnowledge
    const float bn_rs = rsqrtf(1.0f + BN_EPS);
#pragma unroll
    for (int j = 0; j < NT; ++j) {
        int n = j * 16 + lr;
        if ((NOUT % 16) != 0 && n >= NOUT) continue;
        float bi = bias ? bias[n] : 0.0f;
        float g = 1.0f, bt = 0.0f;
        if (bn_gamma) { g = bn_gamma[n] * bn_rs; bt = bn_beta[n]; }
#pragma unroll
        for (int r = 0; r < 8; ++r) {
            int m = m0 + r + half * 8;   // C/D layout: VGPR r -> rows r / r+8
            float v = acc[j][r] + bi;
            v = v * g + bt;
            if (RELU) v = fmaxf(v, 0.0f);
            Y[(size_t)m * NOUT + n] = v;
            if (JK_MODE == 1) {
                JK[(size_t)m * K_DIM + n] = v;
            } else if (JK_MODE == 2) {
                size_t ji = (size_t)m * K_DIM + n;
                JK[ji] = fmaxf(JK[ji], v);
            }
        }
    }
}

// ---------------------------------------------------------------------------
extern "C" void kernel_launch(void* const* d_in, const int* in_sizes, int n_in,
                              void* d_out, int out_size, void* d_ws, size_t ws_size,
                              hipStream_t stream) {
    const float* x        = (const float*)d_in[0];
    const int*   ei       = (const int*)d_in[1];   // [2, E]
    const float* fc0_w    = (const float*)d_in[2];
    const float* fc0_b    = (const float*)d_in[3];
    const float* conv_w   = (const float*)d_in[4]; // [L,128,128]
    const float* W_w      = (const float*)d_in[5];
    const float* W_b      = (const float*)d_in[6];
    const float* bn_gamma = (const float*)d_in[7]; // [L+1,128]
    const float* bn_beta  = (const float*)d_in[8];
    const float* fc_out_w = (const float*)d_in[9];
    const float* fc_out_b = (const float*)d_in[10];
    const float* out_w    = (const float*)d_in[11]; // [40,128]
    const float* out_b    = (const float*)d_in[12];
    float* out = (float*)d_out;

    const int N = in_sizes[0] / K_DIM;  // 50000
    const int E = in_sizes[1] / 2;      // 800000
    const int* src = ei;
    const int* dst = ei + E;

    // Workspace carve-up (floats): disq[N] + 4 x [N,128] buffers
    float* disq = (float*)d_ws;
    size_t off = ((size_t)N + 255) & ~(size_t)255;
    float* h   = disq + off;
    float* jk  = h  + (size_t)N * K_DIM;
    float* xw  = jk + (size_t)N * K_DIM;
    float* agg = xw + (size_t)N * K_DIM;

    const int gemm_blocks = (N + 127) / 128;
    const size_t SH128 = (size_t)128 * 132 * sizeof(float); // 67.6 KB
    const size_t SH40  = (size_t)48  * 132 * sizeof(float);

    // ---- degree normalization: disq = rsqrt(1 + indeg) ----
    deg_init_kernel<<<(N + 255) / 256, 256, 0, stream>>>(disq, N);
    deg_count_kernel<<<(E + 255) / 256, 256, 0, stream>>>(dst, disq, E);
    deg_rsqrt_kernel<<<(N + 255) / 256, 256, 0, stream>>>(disq, N);

    // ---- h = relu(bn0(x @ fc0_w^T + fc0_b));  jk = h ----
    gemm_epilogue_kernel<128, true, 1><<<gemm_blocks, 256, SH128, stream>>>(
        x, fc0_w, fc0_b, bn_gamma, bn_beta, h, jk, N);

    // ---- GCN layers ----
    for (int i = 0; i < 2; ++i) {
        // xw = h @ conv_w[i]^T
        gemm_epilogue_kernel<128, false, 0><<<gemm_blocks, 256, SH128, stream>>>(
            h, conv_w + (size_t)i * K_DIM * K_DIM, nullptr, nullptr, nullptr,
            xw, nullptr, N);
        // agg = self-loop term, then scatter edges
        selfloop_init_kernel<<<(N * 32 + 255) / 256, 256, 0, stream>>>(xw, disq, agg, N);
        scatter_kernel<<<(E + 7) / 8, 256, 0, stream>>>(xw, src, dst, disq, agg, E);
        // h = relu(bn[i+1](agg @ W_w^T + W_b));  jk = max(jk, h)
        gemm_epilogue_kernel<128, true, 2><<<gemm_blocks, 256, SH128, stream>>>(
            agg, W_w, W_b, bn_gamma + (size_t)(i + 1) * K_DIM,
            bn_beta + (size_t)(i + 1) * K_DIM, h, jk, N);
    }

    // ---- t = jk @ fc_out_w^T + fc_out_b  (into xw buffer) ----
    gemm_epilogue_kernel<128, false, 0><<<gemm_blocks, 256, SH128, stream>>>(
        jk, fc_out_w, fc_out_b, nullptr, nullptr, xw, nullptr, N);

    // ---- out = t @ out_w^T + out_b  (C = 40) ----
    gemm_epilogue_kernel<40, false, 0><<<gemm_blocks, 256, SH40, stream>>>(
        xw, out_w, out_b, nullptr, nullptr, out, nullptr, N);
}